// YOLOv1Loss_67577015435959
// MI455X (gfx1250) — compile-verified
//
#include <hip/hip_runtime.h>
#include <cstdint>

#define TPB            256
#define CELLS_PER_TILE 256
#define PR_F           30            // floats per cell, predictions
#define GT_F           24            // floats per cell, ground truth
#define NUM_C          20
#define MAIN_BLOCKS    1024
#define LAMB_OBJ       5.0f
#define LAMB_NOOBJ     0.5f

#define PR_VEC4_PER_TILE (CELLS_PER_TILE * PR_F / 4)   // 1920
#define GT_VEC4_PER_TILE (CELLS_PER_TILE * GT_F / 4)   // 1536
#define PR_ISSUE        ((PR_VEC4_PER_TILE + TPB - 1) / TPB)  // 8 (last clamped)
#define GT_ISSUE        (GT_VEC4_PER_TILE / TPB)              // 6 (exact)
#define ASYNC_PER_TILE  (PR_ISSUE + GT_ISSUE)                 // 14 per wave, uniform

typedef __attribute__((address_space(3))) float lds_float;

// 32-bit LDS byte offset of a __shared__ object (generic -> AS3 addrspacecast).
__device__ __forceinline__ uint32_t lds_offset_of(const void* p) {
  return (uint32_t)(uintptr_t)(lds_float*)(p);
}

// One lane moves 16 bytes memory -> LDS via the gfx1250 async copy path
// (ASYNCcnt). NT hint: data is streamed exactly once.
__device__ __forceinline__ void async_copy_b128(uint32_t lds_byte_off,
                                                const float* gbase,
                                                uint32_t byte_off) {
  asm volatile("global_load_async_to_lds_b128 %0, %1, %2 th:TH_LOAD_NT"
               :: "v"(lds_byte_off), "v"(byte_off), "s"(gbase)
               : "memory");
}

__device__ __forceinline__ void wait_async_0() {
  asm volatile("s_wait_asynccnt 0x0" ::: "memory");
}
// Wait until only the most recent tile's copies (ASYNC_PER_TILE per wave)
// remain outstanding: async loads complete in order, so the older tile's
// data is then guaranteed to be in LDS.
__device__ __forceinline__ void wait_async_tile() {
  asm volatile("s_wait_asynccnt 0xe" ::: "memory");   // 0xe == ASYNC_PER_TILE
}

// Issue one tile's worth of async copies; EXACTLY ASYNC_PER_TILE instructions
// per wave (tail lanes clamp to the last float4 -> uniform ASYNCcnt).
__device__ __forceinline__ void issue_tile(uint32_t lds_pr, uint32_t lds_gt,
                                           const float* __restrict__ pr,
                                           const float* __restrict__ gt,
                                           uint32_t pr_base, uint32_t gt_base,
                                           int tid) {
#pragma unroll
  for (int j = 0; j < PR_ISSUE; ++j) {
    uint32_t i = (uint32_t)(tid + j * TPB);
    if (j * TPB + TPB > PR_VEC4_PER_TILE)             // compile-time for tail
      i = (i < PR_VEC4_PER_TILE) ? i : (PR_VEC4_PER_TILE - 1);
    async_copy_b128(lds_pr + i * 16u, pr, pr_base + i * 16u);
  }
#pragma unroll
  for (int j = 0; j < GT_ISSUE; ++j) {
    uint32_t i = (uint32_t)(tid + j * TPB);
    async_copy_b128(lds_gt + i * 16u, gt, gt_base + i * 16u);
  }
}

__device__ __forceinline__ float iou4(const float* g, const float* p) {
  float gAx = g[0] - 0.5f * g[2], gAy = g[1] - 0.5f * g[3];
  float gBx = g[0] + 0.5f * g[2], gBy = g[1] + 0.5f * g[3];
  float pAx = p[0] - 0.5f * p[2], pAy = p[1] - 0.5f * p[3];
  float pBx = p[0] + 0.5f * p[2], pBy = p[1] + 0.5f * p[3];
  float Ax = fmaxf(gAx, pAx), Ay = fmaxf(gAy, pAy);
  float Bx = fminf(gBx, pBx), By = fminf(gBy, pBy);
  float inter = (Bx - Ax) * (By - Ay);
  float gA = fabsf((gBx - gAx) * (gBy - gAy));
  float pA = fabsf((pBx - pAx) * (pBy - pAy));
  float uni = gA + pA - inter;
  float uni_safe = (uni == 0.0f) ? 1.0f : uni;
  bool valid = (inter >= 0.0f) && (pA != 0.0f) &&
               (pAx >= 0.0f) && (pAy >= 0.0f) &&
               (pBx >= 0.0f) && (pBy >= 0.0f);
  return valid ? (inter / uni_safe) : 0.0f;
}

// Loss for one cell; p -> 30 floats, g -> 24 floats (generic pointers: LDS or global).
__device__ __forceinline__ float cell_loss(const float* p, const float* g) {
  float iou1 = iou4(g, p);
  float iou2 = iou4(g, p + 5);
  const float* p5 = (iou1 >= iou2) ? p : (p + 5);

  float dx = g[0] - p5[0];
  float dy = g[1] - p5[1];
  float l = LAMB_OBJ * (dx * dx + dy * dy);
  float dw = __builtin_amdgcn_sqrtf(g[2]) - __builtin_amdgcn_sqrtf(p5[2]);
  float dh = __builtin_amdgcn_sqrtf(g[3]) - __builtin_amdgcn_sqrtf(p5[3]);
  l += LAMB_OBJ * (dw * dw + dh * dh);
  float dc = 1.0f - p5[4];
  l += dc * dc;

  bool any_obj = false;
#pragma unroll
  for (int c = 0; c < NUM_C; ++c) {
    float gv  = g[4 + c];
    float tgt = (gv == 1.0f) ? 1.0f : 0.0f;
    float d   = tgt - p[10 + c];
    l += d * d;
    any_obj = any_obj || (gv != 0.0f);
  }
  float noobj = LAMB_NOOBJ * (p[4] * p[4] + p[9] * p[9]);
  return any_obj ? l : noobj;
}

__global__ __launch_bounds__(TPB) void yolo_loss_main(
    const float* __restrict__ pr, const float* __restrict__ gt,
    float* __restrict__ partials, int num_tiles) {
  // Double-buffered staging: ~112 KB LDS -> 2 workgroups per 320 KB WGP,
  // every resident block keeps async copies in flight while computing.
  __shared__ __align__(16) float s_pr[2][CELLS_PER_TILE * PR_F];
  __shared__ __align__(16) float s_gt[2][CELLS_PER_TILE * GT_F];
  __shared__ float s_red[TPB];

  const int tid = threadIdx.x;
  const uint32_t lds_pr[2] = { lds_offset_of(&s_pr[0][0]), lds_offset_of(&s_pr[1][0]) };
  const uint32_t lds_gt[2] = { lds_offset_of(&s_gt[0][0]), lds_offset_of(&s_gt[1][0]) };

  float acc = 0.0f;
  int tile = blockIdx.x;

  if (tile < num_tiles) {   // prologue: fill buffer 0 with the first tile
    issue_tile(lds_pr[0], lds_gt[0], pr, gt,
               (uint32_t)tile * (uint32_t)(CELLS_PER_TILE * PR_F * 4),
               (uint32_t)tile * (uint32_t)(CELLS_PER_TILE * GT_F * 4), tid);
  }

  int b = 0;
  for (; tile < num_tiles; tile += gridDim.x) {
    const int next = tile + gridDim.x;
    const bool has_next = next < num_tiles;          // block-uniform

    if (has_next) {
      // buf[b^1] was last READ one iteration ago, before the closing
      // barrier of that iteration -> safe to overwrite now.
      issue_tile(lds_pr[b ^ 1], lds_gt[b ^ 1], pr, gt,
                 (uint32_t)next * (uint32_t)(CELLS_PER_TILE * PR_F * 4),
                 (uint32_t)next * (uint32_t)(CELLS_PER_TILE * GT_F * 4), tid);
      wait_async_tile();   // only next tile's 14 copies still outstanding
    } else {
      wait_async_0();
    }
    __syncthreads();       // all waves' copies for buf[b] are in LDS

    acc += cell_loss(&s_pr[b][tid * PR_F], &s_gt[b][tid * GT_F]);

    __syncthreads();       // all waves done reading buf[b]
    b ^= 1;
  }

  // Deterministic block tree-reduction.
  s_red[tid] = acc;
  __syncthreads();
#pragma unroll
  for (int off = TPB / 2; off > 0; off >>= 1) {
    if (tid < off) s_red[tid] += s_red[tid + off];
    __syncthreads();
  }
  if (tid == 0) partials[blockIdx.x] = s_red[0];
}

__global__ __launch_bounds__(TPB) void yolo_loss_final(
    const float* __restrict__ partials, int nparts,
    const float* __restrict__ pr, const float* __restrict__ gt,
    long long rem_start, int rem_count, float invN,
    float* __restrict__ out) {
  __shared__ float s_red[TPB];
  const int tid = threadIdx.x;
  float a = 0.0f;
  for (int i = tid; i < nparts; i += TPB) a += partials[i];
  if (tid < rem_count) {            // leftover cells (none for the 4096x28x28 shape)
    long long c = rem_start + tid;
    a += cell_loss(pr + c * PR_F, gt + c * GT_F);
  }
  s_red[tid] = a;
  __syncthreads();
#pragma unroll
  for (int off = TPB / 2; off > 0; off >>= 1) {
    if (tid < off) s_red[tid] += s_red[tid + off];
    __syncthreads();
  }
  if (tid == 0) out[0] = s_red[0] * invN;
}

extern "C" void kernel_launch(void* const* d_in, const int* in_sizes, int n_in,
                              void* d_out, int out_size, void* d_ws, size_t ws_size,
                              hipStream_t stream) {
  const float* pr = (const float*)d_in[0];   // (N,28,28,30) f32
  const float* gt = (const float*)d_in[1];   // (N,28,28,24) f32
  float* out      = (float*)d_out;
  float* partials = (float*)d_ws;            // MAIN_BLOCKS floats of scratch

  const long long cells = (long long)in_sizes[0] / PR_F;
  const int num_tiles   = (int)(cells / CELLS_PER_TILE);
  const long long rem_start = (long long)num_tiles * CELLS_PER_TILE;
  const int rem_count   = (int)(cells - rem_start);
  const long long N     = cells / (28 * 28);
  const float invN      = 1.0f / (float)N;

  yolo_loss_main<<<MAIN_BLOCKS, TPB, 0, stream>>>(pr, gt, partials, num_tiles);
  yolo_loss_final<<<1, TPB, 0, stream>>>(partials, MAIN_BLOCKS, pr, gt,
                                         rem_start, rem_count, invN, out);
}